// NestedGNN_87205015978449
// MI455X (gfx1250) — compile-verified
//
#include <hip/hip_runtime.h>
#include <hip/hip_bf16.h>

// ---- problem constants (match reference) ----
#define N_IN_NODES     100000
#define NODES_INNER    50
#define PAD_INNER      64       // 50 rows padded to 4 x 16 tiles
#define N_OUT_NODES    2000
#define NODES_OUTER    125
#define PAD_OUTER      128      // 125 rows padded to 8 x 16 tiles
#define B_GRAPHS       16
#define EPG_IN         400      // 800000 inner edges / 2000 graphs (contiguous per graph)
#define EPG_OUT        1000     // 16000 outer edges / 16 graphs (contiguous per graph)
#define IN_FEATS       64
#define OUT_FEATS      64
#define HF             128      // hidden feats
#define NUM_CLASSES    10
#define MERGED_F       (OUT_FEATS + HF)   // 192

typedef __attribute__((ext_vector_type(2))) float v2f;
typedef __attribute__((ext_vector_type(8))) float v8f;

// One wave computes one 16x16 output tile, branch-free (buffers padded):
//   O[mt*16.., nt*16..] = relu( (rowscale .* A) @ W + bias )
// A: LDS, PAD_M x K (rows >= validRows are zero).  rscale: PAD_M entries.
// W: global, K x 128 row-major.  O: LDS, PAD_M x 128.
// f32 WMMA 16x16x4 layout: A lane l -> (M=l&15, K-pair at 2*(l>>4));
//                          B lane l -> (N=l&15, K-pair at 2*(l>>4));
//                          D vgpr r, lane l -> (M = r + 8*(l>>4), N = l&15).
template <int K>
__device__ __forceinline__ void wave_gemm_relu_tile(
    const float* __restrict__ Alds, const float* __restrict__ rscale,
    const float* __restrict__ Wg,   const float* __restrict__ bg,
    float* __restrict__ Olds, int mt, int nt, int lane)
{
    const int half = lane >> 4;        // K-pair select
    const int lid  = lane & 15;
    const int m    = mt * 16 + lid;    // A row for this lane (always in-bounds: padded)
    const int n    = nt * 16 + lid;    // W col for this lane
    const float rs = rscale[m];        // pad rows multiply zeros -> value irrelevant
    const float* Arow = Alds + m * K;

    v8f acc = {};
#pragma unroll
    for (int k = 0; k < K; k += 4) {
        const int ka = k + 2 * half;
        v2f a = *(const v2f*)(Arow + ka);   // one ds_load_b64 (8B-aligned: K,ka even)
        a.x *= rs;
        a.y *= rs;
        v2f b;
        b.x = Wg[ka * HF + n];
        b.y = Wg[(ka + 1) * HF + n];
        acc = __builtin_amdgcn_wmma_f32_16x16x4_f32(
                  false, a, false, b, (short)0, acc, false, false);
    }
    const float bias = bg[n];
#pragma unroll
    for (int r = 0; r < 8; ++r) {       // unconditional: pad rows never read back
        const int row = mt * 16 + r + 8 * half;
        float v = acc[r] + bias;
        Olds[row * HF + n] = v > 0.0f ? v : 0.0f;
    }
}

// ---------------- Kernel 1: one block per inner graph (2000 blocks) --------
// Fuses: degrees -> conv1(msg+GEMM+relu) -> conv2(msg+GEMM+relu) -> mean ->
// concat with out_layer_feat -> merged[g][192] written to workspace.
__global__ void __launch_bounds__(256) inner_fused_kernel(
    const float* __restrict__ in_feat,  const float* __restrict__ inner_ew,
    const float* __restrict__ W1, const float* __restrict__ b1,
    const float* __restrict__ W2, const float* __restrict__ b2,
    const int* __restrict__ isrc, const int* __restrict__ idst,
    const float* __restrict__ out_feat, float* __restrict__ merged)
{
    extern __shared__ float smem[];
    float* bufF = smem;                      // PAD_INNER*128 (feat / h1 / h2)
    float* agg  = bufF + PAD_INNER * HF;     // PAD_INNER*128
    float* rdo  = agg  + PAD_INNER * HF;     // PAD_INNER
    float* rdi  = rdo  + PAD_INNER;          // PAD_INNER

    const int g     = blockIdx.x;
    const int tid   = threadIdx.x;
    const int lane  = tid & 31;
    const int wave  = tid >> 5;
    const int node0 = g * NODES_INNER;
    const int e0    = g * EPG_IN;

    // ---- degrees (shared by both convs) ----
    for (int i = tid; i < PAD_INNER; i += 256) { rdo[i] = 0.0f; rdi[i] = 0.0f; }
    __syncthreads();
    for (int e = tid; e < EPG_IN; e += 256) {
        atomicAdd(&rdo[isrc[e0 + e] - node0], 1.0f);
        atomicAdd(&rdi[idst[e0 + e] - node0], 1.0f);
    }
    __syncthreads();
    for (int i = tid; i < PAD_INNER; i += 256) {
        rdo[i] = rsqrtf(fmaxf(rdo[i], 1.0f));
        rdi[i] = rsqrtf(fmaxf(rdi[i], 1.0f));
    }
    // ---- load input feats (stride 64), zero padded agg (64 x 64) ----
    for (int i = tid; i < NODES_INNER * IN_FEATS; i += 256)
        bufF[i] = in_feat[node0 * IN_FEATS + i];
    for (int i = tid; i < PAD_INNER * IN_FEATS; i += 256) agg[i] = 0.0f;
    __syncthreads();

    // ---- conv1 message pass: agg[d][f] += ew * rsqrt(deg_out[s]) * x[s][f] ----
    for (int idx = tid; idx < EPG_IN * IN_FEATS; idx += 256) {
        const int e = idx >> 6, f = idx & 63;
        const int s = isrc[e0 + e] - node0, d = idst[e0 + e] - node0;
        atomicAdd(&agg[d * IN_FEATS + f],
                  inner_ew[e0 + e] * rdo[s] * bufF[s * IN_FEATS + f]);
    }
    __syncthreads();
    // ---- GEMM1: h1 = relu((agg .* rdi) @ W1 + b1), K=64 -> bufF ----
    for (int job = wave; job < 32; job += 8)
        wave_gemm_relu_tile<IN_FEATS>(agg, rdi, W1, b1, bufF,
                                      job >> 3, job & 7, lane);
    __syncthreads();

    // ---- conv2 message pass (K=128), padded agg (64 x 128) ----
    for (int i = tid; i < PAD_INNER * HF; i += 256) agg[i] = 0.0f;
    __syncthreads();
    for (int idx = tid; idx < EPG_IN * HF; idx += 256) {
        const int e = idx >> 7, f = idx & 127;
        const int s = isrc[e0 + e] - node0, d = idst[e0 + e] - node0;
        atomicAdd(&agg[d * HF + f],
                  inner_ew[e0 + e] * rdo[s] * bufF[s * HF + f]);
    }
    __syncthreads();
    // ---- GEMM2: h2 = relu((agg .* rdi) @ W2 + b2), K=128 -> bufF ----
    for (int job = wave; job < 32; job += 8)
        wave_gemm_relu_tile<HF>(agg, rdi, W2, b2, bufF,
                                job >> 3, job & 7, lane);
    __syncthreads();

    // ---- mean over 50 nodes + concat with outer node feature ----
    if (tid < HF) {
        float s = 0.0f;
        for (int i = 0; i < NODES_INNER; ++i) s += bufF[i * HF + tid];
        merged[g * MERGED_F + OUT_FEATS + tid] = s * (1.0f / (float)NODES_INNER);
    } else if (tid < HF + OUT_FEATS) {
        const int f = tid - HF;
        merged[g * MERGED_F + f] = out_feat[g * OUT_FEATS + f];
    }
}

// ---- one outer conv layer: zero agg -> message pass -> WMMA GEMM+relu ----
template <int K>
__device__ __forceinline__ void outer_layer(
    float* __restrict__ bufF, float* __restrict__ agg,
    const float* __restrict__ rdo, const float* __restrict__ rdi,
    const float* __restrict__ Wg, const float* __restrict__ bg,
    const int* __restrict__ osrc, const int* __restrict__ odst,
    const float* __restrict__ outer_ew,
    int e0, int node0, int tid, int wave, int lane)
{
    for (int i = tid; i < PAD_OUTER * K; i += 256) agg[i] = 0.0f;
    __syncthreads();
    for (int idx = tid; idx < EPG_OUT * K; idx += 256) {
        const int e = idx / K, f = idx % K;
        const int s = osrc[e0 + e] - node0, d = odst[e0 + e] - node0;
        atomicAdd(&agg[d * K + f],
                  outer_ew[e0 + e] * rdo[s] * bufF[s * K + f]);
    }
    __syncthreads();
    for (int job = wave; job < 64; job += 8)   // 8 m-tiles x 8 n-tiles
        wave_gemm_relu_tile<K>(agg, rdi, Wg, bg, bufF, job >> 3, job & 7, lane);
    __syncthreads();
}

// ---------------- Kernel 2: one block per outer graph (16 blocks) ----------
// Fuses 4 graph convs + mean_nodes + classifier.
__global__ void __launch_bounds__(256) outer_fused_kernel(
    const float* __restrict__ merged, const float* __restrict__ outer_ew,
    const float* __restrict__ W3, const float* __restrict__ b3,
    const float* __restrict__ W4, const float* __restrict__ b4,
    const float* __restrict__ W5, const float* __restrict__ b5,
    const float* __restrict__ W6, const float* __restrict__ b6,
    const float* __restrict__ Wc, const float* __restrict__ bc,
    const int* __restrict__ osrc, const int* __restrict__ odst,
    float* __restrict__ out)
{
    extern __shared__ float smem[];
    float* bufF = smem;                          // PAD_OUTER*192 (feat / h)
    float* agg  = bufF + PAD_OUTER * MERGED_F;   // PAD_OUTER*192
    float* rdo  = agg  + PAD_OUTER * MERGED_F;   // PAD_OUTER
    float* rdi  = rdo  + PAD_OUTER;              // PAD_OUTER
    float* hg   = rdi  + PAD_OUTER;              // 128

    const int g     = blockIdx.x;
    const int tid   = threadIdx.x;
    const int lane  = tid & 31;
    const int wave  = tid >> 5;
    const int node0 = g * NODES_OUTER;
    const int e0    = g * EPG_OUT;

    for (int i = tid; i < PAD_OUTER; i += 256) { rdo[i] = 0.0f; rdi[i] = 0.0f; }
    __syncthreads();
    for (int e = tid; e < EPG_OUT; e += 256) {
        atomicAdd(&rdo[osrc[e0 + e] - node0], 1.0f);
        atomicAdd(&rdi[odst[e0 + e] - node0], 1.0f);
    }
    __syncthreads();
    for (int i = tid; i < PAD_OUTER; i += 256) {
        rdo[i] = rsqrtf(fmaxf(rdo[i], 1.0f));
        rdi[i] = rsqrtf(fmaxf(rdi[i], 1.0f));
    }
    for (int i = tid; i < NODES_OUTER * MERGED_F; i += 256)
        bufF[i] = merged[node0 * MERGED_F + i];
    __syncthreads();

    outer_layer<MERGED_F>(bufF, agg, rdo, rdi, W3, b3, osrc, odst, outer_ew,
                          e0, node0, tid, wave, lane);
    outer_layer<HF>(bufF, agg, rdo, rdi, W4, b4, osrc, odst, outer_ew,
                    e0, node0, tid, wave, lane);
    outer_layer<HF>(bufF, agg, rdo, rdi, W5, b5, osrc, odst, outer_ew,
                    e0, node0, tid, wave, lane);
    outer_layer<HF>(bufF, agg, rdo, rdi, W6, b6, osrc, odst, outer_ew,
                    e0, node0, tid, wave, lane);

    // ---- mean over 125 nodes ----
    if (tid < HF) {
        float s = 0.0f;
        for (int i = 0; i < NODES_OUTER; ++i) s += bufF[i * HF + tid];
        hg[tid] = s * (1.0f / (float)NODES_OUTER);
    }
    __syncthreads();
    // ---- classifier: out[g][c] = hg @ Wc[:,c] + bc[c] ----
    if (tid < NUM_CLASSES) {
        float s = bc[tid];
        for (int f = 0; f < HF; ++f) s += hg[f] * Wc[f * NUM_CLASSES + tid];
        out[g * NUM_CLASSES + tid] = s;
    }
}

extern "C" void kernel_launch(void* const* d_in, const int* in_sizes, int n_in,
                              void* d_out, int out_size, void* d_ws, size_t ws_size,
                              hipStream_t stream) {
    (void)in_sizes; (void)n_in; (void)out_size; (void)ws_size;
    const float* in_feat  = (const float*)d_in[0];
    const float* out_feat = (const float*)d_in[1];
    const float* inner_ew = (const float*)d_in[2];
    const float* outer_ew = (const float*)d_in[3];
    const float* W1 = (const float*)d_in[4];  const float* b1 = (const float*)d_in[5];
    const float* W2 = (const float*)d_in[6];  const float* b2 = (const float*)d_in[7];
    const float* W3 = (const float*)d_in[8];  const float* b3 = (const float*)d_in[9];
    const float* W4 = (const float*)d_in[10]; const float* b4 = (const float*)d_in[11];
    const float* W5 = (const float*)d_in[12]; const float* b5 = (const float*)d_in[13];
    const float* W6 = (const float*)d_in[14]; const float* b6 = (const float*)d_in[15];
    const float* Wc = (const float*)d_in[16]; const float* bc = (const float*)d_in[17];
    const int* isrc = (const int*)d_in[18];   const int* idst = (const int*)d_in[19];
    const int* osrc = (const int*)d_in[20];   const int* odst = (const int*)d_in[21];
    float* merged = (float*)d_ws;             // 2000 x 192 fp32 = 1.5 MB scratch
    float* out    = (float*)d_out;            // 16 x 10 fp32

    const size_t shm1 = (size_t)(PAD_INNER * HF * 2 + PAD_INNER * 2) * sizeof(float);
    const size_t shm2 = (size_t)(PAD_OUTER * MERGED_F * 2 + PAD_OUTER * 2 + HF) * sizeof(float);

    inner_fused_kernel<<<N_OUT_NODES, 256, shm1, stream>>>(
        in_feat, inner_ew, W1, b1, W2, b2, isrc, idst, out_feat, merged);
    outer_fused_kernel<<<B_GRAPHS, 256, shm2, stream>>>(
        merged, outer_ew, W3, b3, W4, b4, W5, b5, W6, b6, Wc, bc,
        osrc, odst, out);
}